// GAE_43104291783484
// MI455X (gfx1250) — compile-verified
//
#include <hip/hip_runtime.h>
#include <math.h>

typedef __attribute__((ext_vector_type(2))) float v2f;
typedef __attribute__((ext_vector_type(4))) float v4f;
typedef __attribute__((ext_vector_type(8))) float v8f;

#define IN_CH 512
#define OUT_CH 128

// ---------------------------------------------------------------------------
// Degree / normalization kernels
// ---------------------------------------------------------------------------
__global__ void init_deg_kernel(float* __restrict__ deg, int n) {
    int i = blockIdx.x * blockDim.x + threadIdx.x;
    if (i < n) deg[i] = 1.0f;  // self-loop contributes 1 to every node
}

__global__ void accum_deg_kernel(const long long* __restrict__ dst,
                                 float* __restrict__ deg, int e) {
    int i = blockIdx.x * blockDim.x + threadIdx.x;
    if (i < e) atomicAdd(&deg[dst[i]], 1.0f);
}

__global__ void rsqrt_kernel(float* __restrict__ deg, int n) {
    int i = blockIdx.x * blockDim.x + threadIdx.x;
    if (i < n) deg[i] = rsqrtf(deg[i]);  // deg >= 1 always (self loops)
}

// ---------------------------------------------------------------------------
// Encoder GEMM: xw = x @ W_enc   [M,512] x [512,128] -> [M,128]
// One wave = one 16x16 C tile across 8 N-tiles (full N=128).
// W_enc staged in LDS in 128-row K-chunks, K-PAIR PACKED:
//   Bp[kp*128 + col] = { W[2kp][col], W[2kp+1][col] }   (float2)
// so each WMMA B fragment is one aligned ds_load_b64 into an even VGPR pair.
// ---------------------------------------------------------------------------
__global__ __launch_bounds__(512) void gemm_enc_kernel(
    const float* __restrict__ x, const float* __restrict__ W,
    float* __restrict__ xw, int M) {
    __shared__ v2f Bp[64 * OUT_CH];  // 64 KB
    const int tid    = threadIdx.x;
    const int lane   = tid & 31;
    const int wave   = tid >> 5;
    const int mtiles = M >> 4;
    const int mt     = blockIdx.x * 16 + wave;
    const bool valid = mt < mtiles;
    const int col    = lane & 15;
    const int klane  = (lane >> 4) << 1;  // 0 or 2 (half-wave K split, always even)
    const float* __restrict__ A = x + (size_t)((valid ? mt : 0) * 16 + col) * IN_CH;

    v8f acc[8];
#pragma unroll
    for (int nt = 0; nt < 8; ++nt) acc[nt] = v8f{};

    for (int kc = 0; kc < IN_CH; kc += 128) {
        __syncthreads();
        const float* src = W + (size_t)kc * OUT_CH;
#pragma unroll
        for (int idx = tid; idx < 64 * OUT_CH; idx += 512) {
            int kp = idx >> 7, c = idx & 127;
            Bp[idx] = v2f{src[(2 * kp + 0) * OUT_CH + c],
                          src[(2 * kp + 1) * OUT_CH + c]};
        }
        __syncthreads();
#pragma unroll 4
        for (int ks = 0; ks < 128; ks += 4) {
            const int k = ks + klane;  // even
            v2f a = *(const v2f*)(A + kc + k);
            const v2f* brow = &Bp[(k >> 1) * OUT_CH + col];
#pragma unroll
            for (int nt = 0; nt < 8; ++nt) {
                v2f b = brow[nt * 16];
                acc[nt] = __builtin_amdgcn_wmma_f32_16x16x4_f32(
                    false, a, false, b, (short)0, acc[nt], false, false);
            }
        }
    }
    if (valid) {
        const int rbase = mt * 16 + ((lane >> 4) << 3);
#pragma unroll
        for (int nt = 0; nt < 8; ++nt)
#pragma unroll
            for (int r = 0; r < 8; ++r)
                xw[(size_t)(rbase + r) * OUT_CH + nt * 16 + col] = acc[nt][r];
    }
}

// ---------------------------------------------------------------------------
// z init with self-loop term: z[i,:] = xw[i,:] * dinv[i]^2
// ---------------------------------------------------------------------------
__global__ void zinit_kernel(const float* __restrict__ xw,
                             const float* __restrict__ dinv,
                             float* __restrict__ z, int n) {
    int t = blockIdx.x * blockDim.x + threadIdx.x;
    int node = t >> 5, c = (t & 31) * 4;
    if (node < n) {
        float s = dinv[node];
        s *= s;
        v4f v = *(const v4f*)(xw + (size_t)node * OUT_CH + c);
        *(v4f*)(z + (size_t)node * OUT_CH + c) =
            v4f{v[0] * s, v[1] * s, v[2] * s, v[3] * s};
    }
}

// ---------------------------------------------------------------------------
// Edge scatter: one wave per edge, one float4 per lane (32*4 = 128 feats).
// xw and z (25.6 MB each) fit the 192 MB L2 -> atomics resolve in L2.
// ---------------------------------------------------------------------------
__global__ __launch_bounds__(256) void edge_scatter_kernel(
    const long long* __restrict__ src, const long long* __restrict__ dst,
    const float* __restrict__ dinv, const float* __restrict__ xw,
    float* __restrict__ z, int e) {
    int w = (blockIdx.x * 256 + threadIdx.x) >> 5;
    int lane = threadIdx.x & 31;
    if (w >= e) return;
    long long s = src[w], d = dst[w];
    float c = dinv[s] * dinv[d];
    v4f v = *(const v4f*)(xw + (size_t)s * OUT_CH + lane * 4);
    float* zd = z + (size_t)d * OUT_CH + lane * 4;
    atomicAdd(zd + 0, v[0] * c);
    atomicAdd(zd + 1, v[1] * c);
    atomicAdd(zd + 2, v[2] * c);
    atomicAdd(zd + 3, v[3] * c);
}

// ---------------------------------------------------------------------------
// Decoder GEMM with fused bias+relu on A:
// out = relu(z + b) @ W_dec   [M,128] x [128,512] -> [M,512]
// Full K=128 A-strip kept in 32 v2f registers, reused for all 32 N-tiles.
// W_dec staged in LDS in 128-column N-chunks, K-pair packed (as above).
// ---------------------------------------------------------------------------
__global__ __launch_bounds__(512) void gemm_dec_kernel(
    const float* __restrict__ z, const float* __restrict__ bias,
    const float* __restrict__ W, float* __restrict__ out, int M) {
    __shared__ v2f Bp[64 * 128];  // 64 KB
    const int tid    = threadIdx.x;
    const int lane   = tid & 31;
    const int wave   = tid >> 5;
    const int mtiles = M >> 4;
    const int mt     = blockIdx.x * 16 + wave;
    const bool valid = mt < mtiles;
    const int col    = lane & 15;
    const int klane  = (lane >> 4) << 1;
    const float* __restrict__ Z = z + (size_t)((valid ? mt : 0) * 16 + col) * OUT_CH;

    v2f afrag[32];
#pragma unroll
    for (int ks = 0; ks < 32; ++ks) {
        int k = ks * 4 + klane;
        float a0 = Z[k + 0] + bias[k + 0];
        float a1 = Z[k + 1] + bias[k + 1];
        afrag[ks] = v2f{fmaxf(a0, 0.0f), fmaxf(a1, 0.0f)};
    }

    for (int nc = 0; nc < IN_CH; nc += 128) {
        __syncthreads();
#pragma unroll
        for (int idx = tid; idx < 64 * 128; idx += 512) {
            int kp = idx >> 7, c = idx & 127;
            Bp[idx] = v2f{W[(size_t)(2 * kp + 0) * IN_CH + nc + c],
                          W[(size_t)(2 * kp + 1) * IN_CH + nc + c]};
        }
        __syncthreads();
#pragma unroll
        for (int nt = 0; nt < 8; ++nt) {
            v8f acc = v8f{};
            const v2f* bcol = &Bp[nt * 16 + col];
#pragma unroll 8
            for (int ks = 0; ks < 32; ++ks) {
                int k = ks * 4 + klane;  // even
                v2f b = bcol[(k >> 1) * 128];
                acc = __builtin_amdgcn_wmma_f32_16x16x4_f32(
                    false, afrag[ks], false, b, (short)0, acc, false, false);
            }
            if (valid) {
                const int rbase = mt * 16 + ((lane >> 4) << 3);
#pragma unroll
                for (int r = 0; r < 8; ++r)
                    out[(size_t)(rbase + r) * IN_CH + nc + nt * 16 + col] = acc[r];
            }
        }
    }
}

// ---------------------------------------------------------------------------
// Row softmax over 512 features, in place. One wave per row (16 vals/lane).
// ---------------------------------------------------------------------------
__global__ __launch_bounds__(256) void softmax_kernel(float* __restrict__ out, int M) {
    int row = (blockIdx.x * 256 + threadIdx.x) >> 5;
    int lane = threadIdx.x & 31;
    if (row >= M) return;
    float* p = out + (size_t)row * IN_CH;
    v4f v[4];
    float mx = -INFINITY;
#pragma unroll
    for (int i = 0; i < 4; ++i) {
        v[i] = *(const v4f*)(p + (lane + i * 32) * 4);
        mx = fmaxf(mx, fmaxf(fmaxf(v[i][0], v[i][1]), fmaxf(v[i][2], v[i][3])));
    }
#pragma unroll
    for (int off = 16; off > 0; off >>= 1)
        mx = fmaxf(mx, __shfl_xor(mx, off, 32));
    float sum = 0.0f;
#pragma unroll
    for (int i = 0; i < 4; ++i) {
#pragma unroll
        for (int j = 0; j < 4; ++j) {
            float e = __expf(v[i][j] - mx);
            v[i][j] = e;
            sum += e;
        }
    }
#pragma unroll
    for (int off = 16; off > 0; off >>= 1)
        sum += __shfl_xor(sum, off, 32);
    float inv = 1.0f / sum;
#pragma unroll
    for (int i = 0; i < 4; ++i) {
        *(v4f*)(p + (lane + i * 32) * 4) =
            v4f{v[i][0] * inv, v[i][1] * inv, v[i][2] * inv, v[i][3] * inv};
    }
}

// ---------------------------------------------------------------------------
extern "C" void kernel_launch(void* const* d_in, const int* in_sizes, int n_in,
                              void* d_out, int out_size, void* d_ws, size_t ws_size,
                              hipStream_t stream) {
    const float* x        = (const float*)d_in[0];
    const long long* eidx = (const long long*)d_in[1];
    const float* W_enc    = (const float*)d_in[2];
    const float* b_enc    = (const float*)d_in[3];
    const float* W_dec    = (const float*)d_in[4];
    float* out            = (float*)d_out;

    const int N = in_sizes[0] / IN_CH;   // 50000 nodes
    const int E = in_sizes[1] / 2;       // 1.6M edges
    const long long* src = eidx;
    const long long* dst = eidx + E;

    // workspace layout: [deg/dinv (N f32)] [xw (N*128 f32)] [z (N*128 f32)]
    size_t o1 = ((size_t)N * 4 + 511) & ~(size_t)511;
    float* deg = (float*)d_ws;  // becomes dinv in place
    float* xw  = (float*)((char*)d_ws + o1);
    float* z   = xw + (size_t)N * OUT_CH;

    const int mtiles  = N / 16;              // 3125 (N is multiple of 16)
    const int gblocks = (mtiles + 15) / 16;  // 196 blocks of 16 waves

    init_deg_kernel<<<(N + 255) / 256, 256, 0, stream>>>(deg, N);
    accum_deg_kernel<<<(E + 255) / 256, 256, 0, stream>>>(dst, deg, E);
    rsqrt_kernel<<<(N + 255) / 256, 256, 0, stream>>>(deg, N);

    gemm_enc_kernel<<<gblocks, 512, 0, stream>>>(x, W_enc, xw, N);

    zinit_kernel<<<((size_t)N * 32 + 255) / 256, 256, 0, stream>>>(xw, deg, z, N);
    edge_scatter_kernel<<<((size_t)E * 32 + 255) / 256, 256, 0, stream>>>(
        src, dst, deg, xw, z, E);

    gemm_dec_kernel<<<gblocks, 512, 0, stream>>>(z, b_enc, W_dec, out, N);
    softmax_kernel<<<(N * 32 + 255) / 256, 256, 0, stream>>>(out, N);
}